// HelixMemory_10634339025369
// MI455X (gfx1250) — compile-verified
//
#include <hip/hip_runtime.h>
#include <stdint.h>

// Problem constants (match reference): B=8, S=1024, D=2048, M=2048, N=9, C=2
#define NBATCH    8
#define DCOL      2048
#define MEMROWS   2558      // M + N_CONV = 2048 + 510
#define OUTROWS   2558
#define COPYROWS  2048      // short_memory rows -> out rows 510..2557
#define POOLROWS  510       // helix parts -> out rows 0..509

typedef float v2f __attribute__((ext_vector_type(2)));
typedef float v8f __attribute__((ext_vector_type(8)));

// ---------------------------------------------------------------------------
// Kernel 1: bulk copy of short_memory (out rows 510..2557).
// Per block: one output row (8192 B). Each thread moves 2x16B chunks through
// LDS via the CDNA5 async path (ASYNCcnt-tracked, no VGPR data movement).
// ---------------------------------------------------------------------------
__global__ __launch_bounds__(256) void helix_copy(const float* __restrict__ inputs,
                                                  const float* __restrict__ memory,
                                                  float* __restrict__ out) {
  __shared__ __align__(16) float smem[DCOL];   // 8 KB bounce buffer
  const int t   = threadIdx.x;
  const int bid = blockIdx.x;
  const int j   = bid & (COPYROWS - 1);        // row within short_memory
  const int b   = bid >> 11;

  // short_memory row j = new_memory[512+j] = memory[512+j] (j<2046) else inputs[j-2046]
  const float* src = (j < 2046)
      ? memory + ((size_t)b * MEMROWS + 512 + j) * DCOL
      : inputs + ((size_t)b * 1024 + (j - 2046)) * DCOL;
  float* dst = out + ((size_t)b * OUTROWS + 510 + j) * DCOL;

  uint32_t lds0 = (uint32_t)(uintptr_t)(void*)&smem[t * 4]; // LDS byte offset
  uint32_t lds1 = lds0 + 256 * 16;
  uint64_t g0 = (uint64_t)(uintptr_t)(const void*)(src + t * 4);
  uint64_t g1 = g0 + 256 * 16;

  asm volatile("global_load_async_to_lds_b128 %0, %1, off"
               :: "v"(lds0), "v"(g0) : "memory");
  asm volatile("global_load_async_to_lds_b128 %0, %1, off"
               :: "v"(lds1), "v"(g1) : "memory");
  asm volatile("s_wait_asynccnt 0x0" ::: "memory");          // loads landed in LDS

  uint64_t d0 = (uint64_t)(uintptr_t)(void*)(dst + t * 4);
  uint64_t d1 = d0 + 256 * 16;
  asm volatile("global_store_async_from_lds_b128 %0, %1, off"
               :: "v"(d0), "v"(lds0) : "memory");
  asm volatile("global_store_async_from_lds_b128 %0, %1, off"
               :: "v"(d1), "v"(lds1) : "memory");
  // s_endpgm performs implicit S_WAIT_IDLE -> async stores drain.
}

// ---------------------------------------------------------------------------
// Kernel 2: helix parts (out rows 0..509) as WMMA weighted row-sums.
// out[m, n] = inv_m * sum_{k in [start_m, start_m+cnt_m)} memory[k, n]
// D(16x16) += A(16x4 weights) x B(4x16 memory rows), V_WMMA_F32_16X16X4_F32.
// One wave per (batch, 16-row tile, 16-col tile); 8 waves / block.
// K-span is made wave-uniform via readfirstlane so the loop is scalar
// (s_cmp/s_cbranch) and EXEC is provably all-1s at every WMMA.
// ---------------------------------------------------------------------------
__global__ __launch_bounds__(256) void helix_pool(const float* __restrict__ memory,
                                                  float* __restrict__ out) {
  const int lane = threadIdx.x & 31;
  const int wave = threadIdx.x >> 5;
  const int bid  = blockIdx.x;
  const int colGroup = bid & 15;
  const int rowTile  = (bid >> 4) & 31;
  const int b        = bid >> 9;
  const int c0 = (colGroup * 8 + wave) * 16;   // 128 col tiles of 16
  const int r0 = rowTile * 16;

  // Per-output-row (m) averaging run over memory rows: derived closed form.
  const int m = r0 + (lane & 15);
  int startRow = 1536, cnt = 0;
  float inv = 0.0f;
  if (m < POOLROWS) {
    const int u    = m + 2;                        // in [2, 511]
    const int p    = 31 - __builtin_clz((unsigned)u);
    const int off  = u - (1 << p);
    const int half = 1 << (p - 1);
    if (off < half) { cnt = 1 << (9 - p);  startRow = 1280 + cnt * off; }
    else            { cnt = 1 << (10 - p); startRow = 1024 + cnt * (off - half); }
    inv = 1.0f / (float)cnt;
  }
  const int endRow = startRow + cnt;

  // Wave-wide K-span [lo, hi) via wave32 butterfly reduction, then force
  // it into SGPRs: scalar loop, full EXEC through the WMMA body.
  int lo = (m < POOLROWS) ? startRow : 1536;
  int hi = (m < POOLROWS) ? endRow   : 1024;
  #pragma unroll
  for (int s = 16; s >= 1; s >>= 1) {
    int olo = __shfl_xor(lo, s, 32);
    int ohi = __shfl_xor(hi, s, 32);
    lo = (olo < lo) ? olo : lo;
    hi = (ohi > hi) ? ohi : hi;
  }
  lo = __builtin_amdgcn_readfirstlane(lo);
  hi = __builtin_amdgcn_readfirstlane(hi);

  const float* memB = memory + (size_t)b * MEMROWS * DCOL;
  const int col  = c0 + (lane & 15);           // B: N = lane&15
  const int koff = (lane >> 4) * 2;            // A/B: K slot pair per half-wave
  const float* rowPtr = memB + (size_t)(lo + koff) * DCOL + col;

  v8f acc = {0.f, 0.f, 0.f, 0.f, 0.f, 0.f, 0.f, 0.f};
  for (int kb = lo; kb < hi; kb += 4) {
    const int kr0 = kb + koff;
    const int kr1 = kr0 + 1;
    v2f a, bv;
    a.x = (kr0 >= startRow && kr0 < endRow) ? inv : 0.0f;
    a.y = (kr1 >= startRow && kr1 < endRow) ? inv : 0.0f;
    bv.x = rowPtr[0];                          // row kr0   (reads < row 1540 << 2558)
    bv.y = rowPtr[DCOL];                       // row kr0+1
    rowPtr += 4 * DCOL;
    // D = A x B + C   (8 args: neg_a, A, neg_b, B, c_mod, C, reuse_a, reuse_b)
    acc = __builtin_amdgcn_wmma_f32_16x16x4_f32(
        false, a, false, bv, (short)0, acc, false, false);
  }

  // D layout: VGPR i -> M = i (lanes 0-15) / M = 8+i (lanes 16-31), N = lane&15
  float* outB = out + (size_t)b * OUTROWS * DCOL;
  const int rbase = r0 + ((lane >> 4) ? 8 : 0);
  float* op = outB + (size_t)rbase * DCOL + col;
  if (r0 + 16 <= POOLROWS) {                   // common case: straight 8-store clause
    #pragma unroll
    for (int i = 0; i < 8; ++i) op[(size_t)i * DCOL] = acc[i];
  } else {                                     // last row-tile: rows 510/511 don't exist
    #pragma unroll
    for (int i = 0; i < 8; ++i) {
      if (rbase + i < POOLROWS) op[(size_t)i * DCOL] = acc[i];
    }
  }
}

// ---------------------------------------------------------------------------
extern "C" void kernel_launch(void* const* d_in, const int* in_sizes, int n_in,
                              void* d_out, int out_size, void* d_ws, size_t ws_size,
                              hipStream_t stream) {
  (void)in_sizes; (void)n_in; (void)out_size; (void)d_ws; (void)ws_size;
  const float* inputs = (const float*)d_in[0];   // (8, 1024, 2048) f32
  const float* memory = (const float*)d_in[1];   // (8, 2558, 2048) f32
  float* out = (float*)d_out;                    // (8, 2558, 2048) f32

  // Copy first: warms L2 with memory rows 512..2557; pool kernel re-reads
  // rows 1024..1535 (32 MB total) from the 192 MB L2.
  helix_copy<<<dim3(NBATCH * COPYROWS), dim3(256), 0, stream>>>(inputs, memory, out);
  helix_pool<<<dim3(NBATCH * 32 * 16), dim3(256), 0, stream>>>(memory, out);
}